// SimpleGatedBasisGcn_37014028157514
// MI455X (gfx1250) — compile-verified
//
#include <hip/hip_runtime.h>

typedef __attribute__((ext_vector_type(2))) float v2f;
typedef __attribute__((ext_vector_type(8))) float v8f;

#define DIM 64

// ---------------------------------------------------------------------------
// Kernel 1: per-node dual GEMM with f32 WMMA.
//   P = relu(NC @ Vp + bm),  G = NC @ Vg + bg
// One wave computes a 16-row x 64-col tile of BOTH outputs (A-frag shared).
// V_WMMA_F32_16X16X4_F32:
//   A 16x4:  lanes 0-15 -> M=lane, K={k0,k0+1}; lanes 16-31 -> K={k0+2,k0+3}
//   B 4x16:  lanes 0-15 -> N=lane, K={k0,k0+1}; lanes 16-31 -> K={k0+2,k0+3}
//   C 16x16: VGPR i -> M = i + 8*(lane>=16), N = lane&15
// ---------------------------------------------------------------------------
__global__ __launch_bounds__(128) void node_proj_wmma(
    const float* __restrict__ nc,   // [N,64]
    const float* __restrict__ Vp,   // [64,64]
    const float* __restrict__ Vg,   // [64,64]
    const float* __restrict__ bm,   // [64]
    const float* __restrict__ bg,   // [64]
    float* __restrict__ P,          // [N,64]
    float* __restrict__ G,          // [N,64]
    int N)
{
    const int lane = threadIdx.x & 31;
    const int wave = threadIdx.x >> 5;
    const int tileRow = (blockIdx.x * 4 + wave) * 16;

    const int m  = lane & 15;         // row within tile (A) / col within n-tile (B,C)
    const int hk = (lane >> 4) * 2;   // k sub-offset 0 or 2

    int rowA = tileRow + m;           // clamp OOB rows for loads (EXEC must stay full
    if (rowA > N - 1) rowA = N - 1;   // for WMMA); stores are guarded per-lane below
    const float* arow = nc + (size_t)rowA * DIM;

    for (int n0 = 0; n0 < DIM; n0 += 16) {
        const int n = n0 + m;
        v8f accP = {};
        v8f accG = {};
        #pragma unroll
        for (int k0 = 0; k0 < DIM; k0 += 4) {
            const int ka = k0 + hk;
            v2f a, bp, bgf;
            a.x   = arow[ka];
            a.y   = arow[ka + 1];
            bp.x  = Vp[ka * DIM + n];
            bp.y  = Vp[(ka + 1) * DIM + n];
            bgf.x = Vg[ka * DIM + n];
            bgf.y = Vg[(ka + 1) * DIM + n];
            accP = __builtin_amdgcn_wmma_f32_16x16x4_f32(
                false, a, false, bp,  (short)0, accP, false, false);
            accG = __builtin_amdgcn_wmma_f32_16x16x4_f32(
                false, a, false, bgf, (short)0, accG, false, false);
        }
        const float biasM = bm[n];
        const float biasG = bg[n];
        const int rbase = tileRow + (lane >> 4) * 8;
        #pragma unroll
        for (int i = 0; i < 8; ++i) {
            const int r = rbase + i;
            if (r < N) {
                float pv = accP[i] + biasM;
                P[(size_t)r * DIM + n] = pv > 0.0f ? pv : 0.0f;
                G[(size_t)r * DIM + n] = accG[i] + biasG;
            }
        }
    }
}

// ---------------------------------------------------------------------------
// Kernel 2: zero denom and out (harness poisons them; we own init each call).
// ---------------------------------------------------------------------------
__global__ __launch_bounds__(256) void zero2_f4(float4* __restrict__ a,
                                                float4* __restrict__ b, int n)
{
    int i = blockIdx.x * blockDim.x + threadIdx.x;
    if (i < n) {
        a[i] = make_float4(0.f, 0.f, 0.f, 0.f);
        b[i] = make_float4(0.f, 0.f, 0.f, 0.f);
    }
}

// ---------------------------------------------------------------------------
// Kernel 3: denom[r,d] += exp(G[s,d] + Eg[t,d]).  One wave per edge, float2/lane.
// ---------------------------------------------------------------------------
__global__ __launch_bounds__(256) void edge_denom(
    const float* __restrict__ G, const float* __restrict__ Eg,
    const int* __restrict__ snd, const int* __restrict__ rcv,
    const int* __restrict__ typ,
    float* __restrict__ denom, int E)
{
    const int t = blockIdx.x * 256 + threadIdx.x;
    const int e = t >> 5;
    if (e >= E) return;
    const int j = (t & 31) * 2;
    const int s  = snd[e];
    const int r  = rcv[e];
    const int ty = typ[e];
    const float2 g  = *(const float2*)(G  + (size_t)s  * DIM + j);
    const float2 eg = *(const float2*)(Eg + (size_t)ty * DIM + j);
    const float ex0 = __expf(g.x + eg.x);
    const float ex1 = __expf(g.y + eg.y);
    atomicAdd(denom + (size_t)r * DIM + j,     ex0);
    atomicAdd(denom + (size_t)r * DIM + j + 1, ex1);
}

// ---------------------------------------------------------------------------
// Kernel 4: out[r,d] += P[s,d] * exp(G[s,d]+Eg[t,d]) / denom[r,d]
// ---------------------------------------------------------------------------
__global__ __launch_bounds__(256) void edge_scatter(
    const float* __restrict__ P, const float* __restrict__ G,
    const float* __restrict__ Eg,
    const int* __restrict__ snd, const int* __restrict__ rcv,
    const int* __restrict__ typ,
    const float* __restrict__ denom,
    float* __restrict__ out, int E)
{
    const int t = blockIdx.x * 256 + threadIdx.x;
    const int e = t >> 5;
    if (e >= E) return;
    const int j = (t & 31) * 2;
    const int s  = snd[e];
    const int r  = rcv[e];
    const int ty = typ[e];
    const float2 g  = *(const float2*)(G  + (size_t)s  * DIM + j);
    const float2 eg = *(const float2*)(Eg + (size_t)ty * DIM + j);
    const float2 p  = *(const float2*)(P  + (size_t)s  * DIM + j);
    const float2 dn = *(const float2*)(denom + (size_t)r * DIM + j);
    const float ex0 = __expf(g.x + eg.x);
    const float ex1 = __expf(g.y + eg.y);
    atomicAdd(out + (size_t)r * DIM + j,     p.x * ex0 / dn.x);
    atomicAdd(out + (size_t)r * DIM + j + 1, p.y * ex1 / dn.y);
}

extern "C" void kernel_launch(void* const* d_in, const int* in_sizes, int n_in,
                              void* d_out, int out_size, void* d_ws, size_t ws_size,
                              hipStream_t stream)
{
    const float* nc = (const float*)d_in[0];   // node_codes      [N,64]
    const float* Vp = (const float*)d_in[1];   // V_proj_sender   [64,64]
    const float* Vg = (const float*)d_in[2];   // V_gate_sender   [64,64]
    const float* Eg = (const float*)d_in[3];   // E_gate          [R,64]
    const float* bm = (const float*)d_in[4];   // B_message       [64]
    const float* bg = (const float*)d_in[5];   // B_gate_pre      [64]
    const int* snd  = (const int*)d_in[6];     // sender_idx      [E]
    const int* rcv  = (const int*)d_in[7];     // receiver_idx    [E]
    const int* typ  = (const int*)d_in[8];     // type_idx        [E]
    float* out = (float*)d_out;                // [N,64]

    const int N = in_sizes[0] / DIM;
    const int E = in_sizes[6];

    // workspace: P | G | denom  (3 * N * 64 floats = ~76.8 MB)
    float* P     = (float*)d_ws;
    float* G     = P + (size_t)N * DIM;
    float* denom = G + (size_t)N * DIM;

    // 1) node-side dual GEMM (WMMA f32)
    const int tiles = (N + 15) / 16;
    const int gemmBlocks = (tiles + 3) / 4;
    node_proj_wmma<<<gemmBlocks, 128, 0, stream>>>(nc, Vp, Vg, bm, bg, P, G, N);

    // 2) zero denom + out
    const int n4 = (N * DIM) / 4;
    zero2_f4<<<(n4 + 255) / 256, 256, 0, stream>>>((float4*)denom, (float4*)out, n4);

    // 3) softmax denominators
    const int edgeBlocks = (E + 7) / 8;   // 8 edges (waves) per 256-thread block
    edge_denom<<<edgeBlocks, 256, 0, stream>>>(G, Eg, snd, rcv, typ, denom, E);

    // 4) gated scatter-add
    edge_scatter<<<edgeBlocks, 256, 0, stream>>>(P, G, Eg, snd, rcv, typ, denom, out, E);
}